// DrugGCN_60945585931025
// MI455X (gfx1250) — compile-verified
//
#include <hip/hip_runtime.h>
#include <hip/hip_bf16.h>
#include <math.h>

// ---------------- problem constants (match reference) ----------------
constexpr int NN   = 100000;   // nodes
constexpr int EE   = 1600000;  // edges (before self loops)
constexpr int GG   = 2000;     // graphs
constexpr int INC  = 9;
constexpr int HID  = 256;
constexpr int OUTC = 128;
constexpr float BN_EPS = 1e-5f;

constexpr long long ET = (long long)EE + NN; // edges incl. self loops

// ---------------- workspace layout (float offsets) ----------------
constexpr long long F_DEG  = 0;                       // NN  (deg -> rsqrt(deg) in place)
constexpr long long F_NORM = F_DEG + NN;              // ET
constexpr long long F_SUM1 = F_NORM + ET;             // 256
constexpr long long F_SQ1  = F_SUM1 + HID;
constexpr long long F_SC1  = F_SQ1  + HID;
constexpr long long F_SH1  = F_SC1  + HID;
constexpr long long F_SUM2 = F_SH1  + HID;            // 128
constexpr long long F_SQ2  = F_SUM2 + OUTC;
constexpr long long F_SC2  = F_SQ2  + OUTC;
constexpr long long F_SH2  = F_SC2  + OUTC;
constexpr long long F_S    = F_SH2  + OUTC;           // GG*128 pooled sums
constexpr long long F_CNT  = F_S + (long long)GG * OUTC; // GG
constexpr long long F_H1   = F_CNT + GG;              // NN*256 ; later reused as agg2 (NN*128)
constexpr long long F_AGG1 = F_H1  + (long long)NN * HID; // NN*256
constexpr long long F_H2   = F_AGG1 + (long long)NN * HID; // NN*128
// total ~ 66.06M floats ~ 264 MB

typedef float v2f __attribute__((ext_vector_type(2)));
typedef float v8f __attribute__((ext_vector_type(8)));

// ---------------- tiny helpers ----------------
__global__ void k_zero(float* __restrict__ p, long long n) {
    long long i = (long long)blockIdx.x * blockDim.x + threadIdx.x;
    long long s = (long long)gridDim.x * blockDim.x;
    for (; i < n; i += s) p[i] = 0.0f;
}

__global__ void k_fill(float* __restrict__ p, long long n, float v) {
    long long i = (long long)blockIdx.x * blockDim.x + threadIdx.x;
    long long s = (long long)gridDim.x * blockDim.x;
    for (; i < n; i += s) p[i] = v;
}

// degree accumulation (self-loop contribution pre-filled as 1.0)
__global__ void k_deg(const int* __restrict__ dst, float* __restrict__ deg) {
    int e = blockIdx.x * blockDim.x + threadIdx.x;
    if (e < EE) unsafeAtomicAdd(&deg[dst[e]], 1.0f);
}

__global__ void k_rsqrt_inplace(float* __restrict__ p, int n) {
    int i = blockIdx.x * blockDim.x + threadIdx.x;
    if (i < n) p[i] = rsqrtf(p[i]);
}

// norm[e] = dis[src]*dis[dst]; self-loop entries e in [EE, ET): dis[i]^2
__global__ void k_norm(const int* __restrict__ src, const int* __restrict__ dst,
                       const float* __restrict__ dis, float* __restrict__ nrm) {
    long long e = (long long)blockIdx.x * blockDim.x + threadIdx.x;
    if (e >= ET) return;
    float v;
    if (e < EE) v = dis[src[e]] * dis[dst[e]];
    else { float d = dis[(int)(e - EE)]; v = d * d; }
    nrm[e] = v;
}

// ---------------- GEMM1: h1 = x @ W1  (K=9, bias dropped: cancels in train BN) ----------------
__global__ void __launch_bounds__(HID) k_gemm1(const float* __restrict__ x,
                                               const float* __restrict__ W1,
                                               float* __restrict__ h1) {
    __shared__ float xs[INC];
    const int i = blockIdx.x;
    const int c = threadIdx.x;
    if (c < INC) xs[c] = x[(long long)i * INC + c];
    __syncthreads();
    float acc = 0.0f;
#pragma unroll
    for (int k = 0; k < INC; ++k) acc += xs[k] * W1[k * HID + c];
    h1[(long long)i * HID + c] = acc;
}

// ---------------- edge scatter: agg[dst] += h[src]*norm (float4 + f32 atomics at L2) ----------
template <int F>
__global__ void __launch_bounds__(256) k_scatter(const float* __restrict__ h,
                                                 const int* __restrict__ src,
                                                 const int* __restrict__ dst,
                                                 const float* __restrict__ nrm,
                                                 float* __restrict__ agg) {
    constexpr int TPE = F / 4;        // threads per edge
    constexpr int EPB = 256 / TPE;    // edges per block
    const int tid = threadIdx.x;
    const long long e = (long long)blockIdx.x * EPB + tid / TPE;
    if (e >= ET) return;
    const int f = (tid % TPE) * 4;
    int s, d;
    if (e < EE) { s = src[e]; d = dst[e]; }
    else        { s = d = (int)(e - EE); }
    const float w = nrm[e];
    const float4 v = *reinterpret_cast<const float4*>(h + (long long)s * F + f);
    float* a = agg + (long long)d * F + f;
    unsafeAtomicAdd(a + 0, v.x * w);
    unsafeAtomicAdd(a + 1, v.y * w);
    unsafeAtomicAdd(a + 2, v.z * w);
    unsafeAtomicAdd(a + 3, v.w * w);
}

// ---------------- BN stats: per-channel sum / sumsq (blockDim == C, coalesced rows) ----------
__global__ void k_stats(const float* __restrict__ h, float* __restrict__ sum,
                        float* __restrict__ sq) {
    const int C = blockDim.x;
    const int c = threadIdx.x;
    float s = 0.0f, q = 0.0f;
    for (int r = blockIdx.x; r < NN; r += gridDim.x) {
        float v = h[(long long)r * C + c];
        s += v; q += v * v;
    }
    unsafeAtomicAdd(&sum[c], s);
    unsafeAtomicAdd(&sq[c],  q);
}

// scale/shift from stats (biased var, as torch train BN)
__global__ void k_bnparams(const float* __restrict__ sum, const float* __restrict__ sq,
                           const float* __restrict__ gamma, const float* __restrict__ beta,
                           float* __restrict__ scale, float* __restrict__ shift) {
    const int c = threadIdx.x;
    const float inv_n = 1.0f / (float)NN;
    const float mean = sum[c] * inv_n;
    const float var  = sq[c] * inv_n - mean * mean;
    const float sc   = gamma[c] * rsqrtf(var + BN_EPS);
    scale[c] = sc;
    shift[c] = beta[c] - mean * sc;
}

// fused BN + ReLU in place (HID channels)
__global__ void __launch_bounds__(HID) k_bn_relu(float* __restrict__ h,
                                                 const float* __restrict__ scale,
                                                 const float* __restrict__ shift) {
    const int c = threadIdx.x;
    const long long idx = (long long)blockIdx.x * HID + c;
    float v = h[idx] * scale[c] + shift[c];
    h[idx] = v > 0.0f ? v : 0.0f;
}

// ---------------- GEMM2: C[100000x128] = A[100000x256] @ B[256x128], full-fp32 WMMA ----------
// One 256-thread block (8 waves) per 16-row M tile; wave w owns N-tile [16w,16w+16).
// A tile staged in LDS with stride 260 (bank-conflict-free 16x4 fragment reads).
// Fragment layouts per CDNA5 ISA 7.12.2:
//   A 16x4 f32 : lanes 0-15 -> M=lane, V0=K0 V1=K1 ; lanes 16-31 -> M=lane-16, V0=K2 V1=K3
//   B 4x16 f32 : lanes 0-15 -> N=lane, V0=K0 V1=K1 ; lanes 16-31 -> N=lane-16, V0=K2 V1=K3
//   C/D 16x16  : VGPR r -> M=r (lanes 0-15, N=lane) / M=r+8 (lanes 16-31, N=lane-16)
__global__ void __launch_bounds__(256) k_gemm2_wmma(const float* __restrict__ A,
                                                    const float* __restrict__ B,
                                                    float* __restrict__ C) {
    __shared__ float lds[16 * 260];
    const int tid = threadIdx.x;
    const long long m0 = (long long)blockIdx.x * 16;

    for (int idx = tid; idx < 16 * 256; idx += 256) {
        const int r = idx >> 8, col = idx & 255;
        lds[r * 260 + col] = A[(m0 + r) * HID + col];
    }
    __syncthreads();

    const int lane  = tid & 31;
    const int wave  = tid >> 5;          // N tile index 0..7
    const int n0    = wave * 16;
    const int mr    = lane & 15;         // M (for A) / N (for B) within tile
    const int khalf = (lane >> 4) << 1;  // 0 or 2

    v8f acc = {};
#pragma unroll 4
    for (int k = 0; k < HID; k += 4) {
        v2f a, b;
        const float* ap = &lds[mr * 260 + k + khalf];
        a.x = ap[0];
        a.y = ap[1];
        const float* bp = &B[(long long)(k + khalf) * OUTC + n0 + mr];
        b.x = bp[0];
        b.y = bp[OUTC];
        acc = __builtin_amdgcn_wmma_f32_16x16x4_f32(false, a, false, b,
                                                    (short)0, acc, false, false);
    }

    const long long mbase = m0 + ((lane >> 4) << 3);
    float* cp = C + mbase * OUTC + n0 + mr;
#pragma unroll
    for (int r = 0; r < 8; ++r) cp[(long long)r * OUTC] = acc[r];
}

// ---------------- pool: S[g] += agg2[i], cnt[g] += 1 ----------------
__global__ void __launch_bounds__(OUTC) k_pool(const float* __restrict__ agg2,
                                               const int* __restrict__ batch,
                                               float* __restrict__ S,
                                               float* __restrict__ cnt) {
    const int i = blockIdx.x;
    const int c = threadIdx.x;
    const int g = batch[i];
    unsafeAtomicAdd(&S[(long long)g * OUTC + c], agg2[(long long)i * OUTC + c]);
    if (c == 0) unsafeAtomicAdd(&cnt[g], 1.0f);
}

// ---------------- final: BN2 affine folded into pooled mean, then L2 normalize -------------
__global__ void __launch_bounds__(OUTC) k_final(const float* __restrict__ S,
                                                const float* __restrict__ cnt,
                                                const float* __restrict__ scale,
                                                const float* __restrict__ shift,
                                                float* __restrict__ out) {
    __shared__ float red[OUTC];
    const int g = blockIdx.x;
    const int c = threadIdx.x;
    float cn = cnt[g];
    cn = cn > 1.0f ? cn : 1.0f;
    const float v = (S[(long long)g * OUTC + c] / cn) * scale[c] + shift[c];
    red[c] = v * v;
    __syncthreads();
    for (int s = OUTC / 2; s > 0; s >>= 1) {
        if (c < s) red[c] += red[c + s];
        __syncthreads();
    }
    float nrm = sqrtf(red[0]);
    nrm = nrm > 1e-12f ? nrm : 1e-12f;
    out[(long long)g * OUTC + c] = v / nrm;
}

// ---------------- host driver ----------------
extern "C" void kernel_launch(void* const* d_in, const int* in_sizes, int n_in,
                              void* d_out, int out_size, void* d_ws, size_t ws_size,
                              hipStream_t stream) {
    (void)in_sizes; (void)n_in; (void)out_size; (void)ws_size;

    const float* x   = (const float*)d_in[0];
    const float* W1  = (const float*)d_in[1];
    /* b1 = d_in[2]: provably cancels inside train-mode BN -> skipped */
    const float* g1  = (const float*)d_in[3];
    const float* be1 = (const float*)d_in[4];
    const float* W2  = (const float*)d_in[5];
    /* b2 = d_in[6]: skipped for the same reason */
    const float* g2  = (const float*)d_in[7];
    const float* be2 = (const float*)d_in[8];
    const int*   ei  = (const int*)d_in[9];
    const int*   bat = (const int*)d_in[10];
    const int* src = ei;
    const int* dst = ei + EE;

    float* ws   = (float*)d_ws;
    float* deg  = ws + F_DEG;     // becomes dis in place
    float* nrm  = ws + F_NORM;
    float* sum1 = ws + F_SUM1;  float* sq1 = ws + F_SQ1;
    float* sc1  = ws + F_SC1;   float* sh1 = ws + F_SH1;
    float* sum2 = ws + F_SUM2;  float* sq2 = ws + F_SQ2;
    float* sc2  = ws + F_SC2;   float* sh2 = ws + F_SH2;
    float* S    = ws + F_S;     float* cnt = ws + F_CNT;
    float* h1   = ws + F_H1;    float* agg2 = ws + F_H1;   // aliased (h1 dead before agg2 lives)
    float* agg1 = ws + F_AGG1;
    float* h2   = ws + F_H2;

    float* out = (float*)d_out;

    // --- zero accumulators (fresh every call; harness does not re-poison) ---
    k_zero<<<2048, 256, 0, stream>>>(ws + F_SUM1, F_H1 - F_SUM1);     // stats + S + cnt
    k_zero<<<4096, 256, 0, stream>>>(agg1, (long long)NN * HID);      // conv1 accumulator

    // --- degree / symmetric norm ---
    k_fill<<<512, 256, 0, stream>>>(deg, NN, 1.0f);                   // self-loop
    k_deg<<<(EE + 255) / 256, 256, 0, stream>>>(dst, deg);
    k_rsqrt_inplace<<<(NN + 255) / 256, 256, 0, stream>>>(deg, NN);   // deg -> dis
    k_norm<<<(int)((ET + 255) / 256), 256, 0, stream>>>(src, dst, deg, nrm);

    // --- layer 1 ---
    k_gemm1<<<NN, HID, 0, stream>>>(x, W1, h1);
    k_scatter<HID><<<(int)((ET + 3) / 4), 256, 0, stream>>>(h1, src, dst, nrm, agg1);
    k_stats<<<512, HID, 0, stream>>>(agg1, sum1, sq1);
    k_bnparams<<<1, HID, 0, stream>>>(sum1, sq1, g1, be1, sc1, sh1);
    k_bn_relu<<<NN, HID, 0, stream>>>(agg1, sc1, sh1);

    // h1 is now dead -> its region becomes the conv2 accumulator
    k_zero<<<4096, 256, 0, stream>>>(agg2, (long long)NN * OUTC);

    // --- layer 2 (WMMA GEMM) ---
    k_gemm2_wmma<<<NN / 16, 256, 0, stream>>>(agg1, W2, h2);
    k_scatter<OUTC><<<(int)((ET + 7) / 8), 256, 0, stream>>>(h2, src, dst, nrm, agg2);
    k_stats<<<512, OUTC, 0, stream>>>(agg2, sum2, sq2);
    k_bnparams<<<1, OUTC, 0, stream>>>(sum2, sq2, g2, be2, sc2, sh2);

    // --- pool + fused BN2-affine + L2 normalize ---
    k_pool<<<NN, OUTC, 0, stream>>>(agg2, bat, S, cnt);
    k_final<<<GG, OUTC, 0, stream>>>(S, cnt, sc2, sh2, out);
}